// LocalAttention_17351667875860
// MI455X (gfx1250) — compile-verified
//
#include <hip/hip_runtime.h>

typedef float v2f __attribute__((ext_vector_type(2)));
typedef float v8f __attribute__((ext_vector_type(8)));

#define HGT   128      // height = width = 128
#define CDIM  256
#define NPER  16384    // tokens per batch

// ---- CDNA5 async global->LDS copy (guarded; falls back to VGPR staging) ----
#if defined(__has_builtin)
#if __has_builtin(__builtin_amdgcn_global_load_async_to_lds_b128) && \
    __has_builtin(__builtin_amdgcn_s_wait_asynccnt)
#define USE_ASYNC_COPY 1
#endif
#endif

#ifdef USE_ASYNC_COPY
typedef int v4i_async __attribute__((vector_size(16)));
typedef __attribute__((address_space(1))) v4i_async* gv4p;   // global int4*
typedef __attribute__((address_space(3))) v4i_async* lv4p;   // LDS int4*
#endif

__device__ __forceinline__ void copy16_to_lds(const float* gsrc, float* ldst)
{
#ifdef USE_ASYNC_COPY
    __builtin_amdgcn_global_load_async_to_lds_b128((gv4p)gsrc, (lv4p)ldst, 0, 0);
#else
    *(float4*)ldst = *(const float4*)gsrc;
#endif
}

__device__ __forceinline__ void copy_lds_wait()
{
#ifdef USE_ASYNC_COPY
    __builtin_amdgcn_s_wait_asynccnt(0);
#endif
}

// -------------------------------------------------------------------------
// Kernel 1: att[p][q] = softmax_q( bias_table[rel_idx(p,q)] )   (64x64)
// -------------------------------------------------------------------------
__global__ __launch_bounds__(64) void att_table_kernel(
    const float* __restrict__ bias_table, float* __restrict__ att)
{
    const int p  = threadIdx.x;          // 0..63
    const int i1 = p >> 3, j1 = p & 7;
    float vals[64];
    float mx = -1e30f;
#pragma unroll
    for (int q = 0; q < 64; ++q) {
        const int i2 = q >> 3, j2 = q & 7;
        const int idx = (i1 - i2 + 7) * 15 + (j1 - j2 + 7);
        const float v = bias_table[idx];
        vals[q] = v;
        mx = fmaxf(mx, v);
    }
    float s = 0.0f;
#pragma unroll
    for (int q = 0; q < 64; ++q) { vals[q] = __expf(vals[q] - mx); s += vals[q]; }
    const float inv = 1.0f / s;
#pragma unroll
    for (int q = 0; q < 64; ++q) att[p * 64 + q] = vals[q] * inv;
}

// -------------------------------------------------------------------------
// Kernel 2: fused  V = x @ W1v + b1v   then   Y_win = att @ V_win
// Block: 128 threads (4 wave32).  Grid: (4 channel tiles, 2048 windows).
// B-operands kept TRANSPOSED in LDS ([n][k]) so each lane's (b0,b1) pair is
// one contiguous 8B ds_load_b64 straight into the WMMA source VGPR pair.
// -------------------------------------------------------------------------
__global__ __launch_bounds__(128) void fused_v_att_kernel(
    const float* __restrict__ x, const float* __restrict__ w1,
    const float* __restrict__ b1, const float* __restrict__ att,
    float* __restrict__ Y)
{
    __shared__ float xs[64][68];    // A tile: 64 tokens x 64 K (row-major)
    __shared__ float wst[64][68];   // B tile transposed: [channel n][K]
    __shared__ float vtT[64][68];   // V tile transposed: [channel n][token q]

    const int tid  = threadIdx.x;
    const int wave = tid >> 5;
    const int lane = tid & 31;
    const int half = lane >> 4;
    const int ln   = lane & 15;

    const int cbase = blockIdx.x * 64;       // channel tile base
    const int win   = blockIdx.y;            // 0..2047
    const int bb    = win >> 8;              // batch
    const int wh    = (win >> 4) & 15;       // window row
    const int ww    = win & 15;              // window col

    // token n for window-local p: n = (wh*8 + p/8)*128 + ww*8 + (p%8)
    const size_t nrow0 = (size_t)bb * NPER + (size_t)(wh * 8) * HGT + (size_t)(ww * 8);

    v8f acc[4];
#pragma unroll
    for (int ct = 0; ct < 4; ++ct)
#pragma unroll
        for (int i = 0; i < 8; ++i) acc[ct][i] = 0.0f;

    // staging assignments: 2 threads per row, 32 floats each
    const int prow = tid >> 1;
    const int koff = (tid & 1) * 32;
    const size_t srow = nrow0 + (size_t)(prow >> 3) * HGT + (size_t)(prow & 7);
    const float* xsrc = x + srow * CDIM;

    for (int k0 = 0; k0 < CDIM; k0 += 64) {
        // ---- stage x chunk: async DMA straight into LDS ----
#pragma unroll
        for (int j = 0; j < 8; ++j)
            copy16_to_lds(xsrc + k0 + koff + j * 4, &xs[prow][koff + j * 4]);

        // ---- stage w1 V-columns (j = 3c+2) transposed: wst[n][k] ----
        {
            const int kk = prow;   // K row this thread reads
            const float* wsrc = w1 + (size_t)(k0 + kk) * (3 * CDIM)
                                   + (size_t)(cbase + koff) * 3 + 2;
#pragma unroll
            for (int j = 0; j < 32; ++j) wst[koff + j][kk] = wsrc[3 * j];
        }
        copy_lds_wait();
        __syncthreads();

        // ---- WMMA f32 16x16x4: 16 k-steps over this 64-K chunk ----
#pragma unroll
        for (int ks = 0; ks < 16; ++ks) {
            const int k4 = ks * 4 + 2 * half;
            const v2f a = *(const v2f*)&xs[16 * wave + ln][k4];
#pragma unroll
            for (int ct = 0; ct < 4; ++ct) {
                const v2f bm = *(const v2f*)&wst[ct * 16 + ln][k4];
                acc[ct] = __builtin_amdgcn_wmma_f32_16x16x4_f32(
                    false, a, false, bm, (short)0, acc[ct], false, false);
            }
        }
        __syncthreads();
    }

    // ---- epilogue: add V bias, park V tile (transposed) in LDS ----
#pragma unroll
    for (int ct = 0; ct < 4; ++ct) {
        const int col = ct * 16 + ln;
        const float bv = b1[(cbase + col) * 3 + 2];
        const int q0 = 16 * wave + 8 * half;       // 8 consecutive tokens
#pragma unroll
        for (int i = 0; i < 8; ++i)
            vtT[col][q0 + i] = acc[ct][i] + bv;
#pragma unroll
        for (int i = 0; i < 8; ++i) acc[ct][i] = 0.0f;
    }
    __syncthreads();

    // ---- phase 2: Y = att(64x64) @ V(64x64), K = 64 tokens ----
    const float* arow = att + (16 * wave + ln) * 64;
#pragma unroll
    for (int ks = 0; ks < 16; ++ks) {
        const int k4 = ks * 4 + 2 * half;
        const v2f a = *(const v2f*)(arow + k4);
#pragma unroll
        for (int ct = 0; ct < 4; ++ct) {
            const v2f bm = *(const v2f*)&vtT[ct * 16 + ln][k4];
            acc[ct] = __builtin_amdgcn_wmma_f32_16x16x4_f32(
                false, a, false, bm, (short)0, acc[ct], false, false);
        }
    }

    // ---- scatter Y back to (B,N,C) token order ----
#pragma unroll
    for (int ct = 0; ct < 4; ++ct) {
        const int col = cbase + ct * 16 + ln;
#pragma unroll
        for (int i = 0; i < 8; ++i) {
            const int p = 16 * wave + i + 8 * half;
            const size_t n = nrow0 + (size_t)(p >> 3) * HGT + (size_t)(p & 7);
            Y[n * CDIM + col] = acc[ct][i];
        }
    }
}

// -------------------------------------------------------------------------
// Kernel 3: out = Y @ w2 + b2    (M=131072, K=256, N=256)
// Block: 128 threads (4 waves); tile 64x64; grid (4, 2048).
// w2 tile kept transposed in LDS for contiguous B-fragment pairs.
// -------------------------------------------------------------------------
__global__ __launch_bounds__(128) void gemm2_kernel(
    const float* __restrict__ Y, const float* __restrict__ w2,
    const float* __restrict__ b2, float* __restrict__ out)
{
    __shared__ float ys[64][68];    // A tile row-major
    __shared__ float wtT[64][68];   // B tile transposed: [n][k]

    const int tid  = threadIdx.x;
    const int wave = tid >> 5;
    const int lane = tid & 31;
    const int half = lane >> 4;
    const int ln   = lane & 15;

    const int    n0 = blockIdx.x * 64;
    const size_t m0 = (size_t)blockIdx.y * 64;

    v8f acc[4];
#pragma unroll
    for (int ct = 0; ct < 4; ++ct)
#pragma unroll
        for (int i = 0; i < 8; ++i) acc[ct][i] = 0.0f;

    const int rr   = tid >> 1;
    const int koff = (tid & 1) * 32;

    for (int k0 = 0; k0 < CDIM; k0 += 64) {
        // ---- stage Y rows: async DMA into LDS ----
#pragma unroll
        for (int j = 0; j < 8; ++j)
            copy16_to_lds(Y + (m0 + rr) * CDIM + k0 + koff + j * 4,
                          &ys[rr][koff + j * 4]);

        // ---- stage w2 rows, transpose-scatter into LDS ----
        {
            const float4* wrow = (const float4*)(w2 + (size_t)(k0 + rr) * CDIM
                                                 + n0 + koff);
#pragma unroll
            for (int j = 0; j < 8; ++j) {
                const float4 v = wrow[j];
                wtT[koff + j * 4 + 0][rr] = v.x;
                wtT[koff + j * 4 + 1][rr] = v.y;
                wtT[koff + j * 4 + 2][rr] = v.z;
                wtT[koff + j * 4 + 3][rr] = v.w;
            }
        }
        copy_lds_wait();
        __syncthreads();

#pragma unroll
        for (int ks = 0; ks < 16; ++ks) {
            const int k4 = ks * 4 + 2 * half;
            const v2f a = *(const v2f*)&ys[16 * wave + ln][k4];
#pragma unroll
            for (int ct = 0; ct < 4; ++ct) {
                const v2f bm = *(const v2f*)&wtT[ct * 16 + ln][k4];
                acc[ct] = __builtin_amdgcn_wmma_f32_16x16x4_f32(
                    false, a, false, bm, (short)0, acc[ct], false, false);
            }
        }
        __syncthreads();
    }

#pragma unroll
    for (int ct = 0; ct < 4; ++ct) {
        const int col = n0 + ct * 16 + ln;
        const float bv = b2[col];
#pragma unroll
        for (int i = 0; i < 8; ++i) {
            const size_t row = m0 + 16 * wave + i + 8 * half;
            out[row * CDIM + col] = acc[ct][i] + bv;
        }
    }
}

// -------------------------------------------------------------------------
extern "C" void kernel_launch(void* const* d_in, const int* in_sizes, int n_in,
                              void* d_out, int out_size, void* d_ws, size_t ws_size,
                              hipStream_t stream)
{
    const float* x  = (const float*)d_in[0];
    const float* w1 = (const float*)d_in[1];
    const float* b1 = (const float*)d_in[2];
    const float* w2 = (const float*)d_in[3];
    const float* b2 = (const float*)d_in[4];
    const float* bt = (const float*)d_in[5];
    float* out = (float*)d_out;

    float* ws  = (float*)d_ws;
    float* att = ws;                 // 64*64 floats (16 KB)
    float* Y   = ws + 4096;          // 8*16384*256 floats (128 MB)

    att_table_kernel<<<1, 64, 0, stream>>>(bt, att);
    fused_v_att_kernel<<<dim3(4, 2048), 128, 0, stream>>>(x, w1, b1, att, Y);
    gemm2_kernel<<<dim3(4, 2048), 128, 0, stream>>>(Y, w2, b2, out);
}